// TextDetectionLoss_65343632441624
// MI455X (gfx1250) — compile-verified
//
#include <hip/hip_runtime.h>
#include <hip/hip_bf16.h>

typedef __attribute__((ext_vector_type(2))) float v2f;
typedef __attribute__((ext_vector_type(4))) float v4f;
typedef __attribute__((ext_vector_type(8))) float v8f;

#define NB      16
#define HH      512
#define WW      512
#define MBOX    64
#define NPTS    5
#define PPTS    (MBOX * NPTS)          // 320
#define NELEM   (NB * HH * WW)         // 4194304
#define NVEC4   (NELEM / 4)            // 1048576
#define NBLK1   1024
#define TPB1    256

// Wave-level reduction of 2 floats/lane across 32 lanes using one exact-f32
// WMMA: A = acc (16x4 f32, 2 VGPRs/lane), B = ones(4x16).
// D[m,n] = sum_k A[m,k]; lane n (n<16) holds D[0..7,n], lane n+16 holds
// D[8..15,n]  =>  sum of own 8 components + shfl_xor(16) = full wave sum.
__device__ __forceinline__ float wave_reduce_wmma(v2f acc) {
    v2f ones; ones.x = 1.0f; ones.y = 1.0f;
    v8f c = {0.f, 0.f, 0.f, 0.f, 0.f, 0.f, 0.f, 0.f};
    v8f d = __builtin_amdgcn_wmma_f32_16x16x4_f32(
        /*neg_a=*/false, acc, /*neg_b=*/false, ones,
        /*c_mod=*/(short)0, c, /*reuse_a=*/false, /*reuse_b=*/false);
    float s = d[0] + d[1] + d[2] + d[3] + d[4] + d[5] + d[6] + d[7];
    s += __shfl_xor(s, 16, 32);
    return s;   // all lanes hold the full wave sum
}

__device__ __forceinline__ float bce_elem(float p, float t) {
    return -(t * __logf(p) + (1.0f - t) * __logf(1.0f - p));
}

// ---------------- Kernel 1: streaming BCE + Dice reductions ----------------
__global__ __launch_bounds__(TPB1)
void text_reduce_kernel(const float* __restrict__ pmap,
                        const float* __restrict__ tmap,
                        float* __restrict__ partials) {
    const v4f* p4 = (const v4f*)pmap;
    const v4f* t4 = (const v4f*)tmap;

    v2f abce; abce.x = 0.f; abce.y = 0.f;
    v2f ainter = abce, ap = abce, at = abce;

    const int stride = gridDim.x * blockDim.x;
    for (int i = blockIdx.x * blockDim.x + threadIdx.x; i < NVEC4; i += stride) {
        v4f pv = p4[i];
        v4f tv = t4[i];
        float b0 = bce_elem(pv.x, tv.x);
        float b1 = bce_elem(pv.y, tv.y);
        float b2 = bce_elem(pv.z, tv.z);
        float b3 = bce_elem(pv.w, tv.w);
        abce.x   += b0 + b1;
        abce.y   += b2 + b3;
        ainter.x += pv.x * tv.x + pv.y * tv.y;
        ainter.y += pv.z * tv.z + pv.w * tv.w;
        ap.x     += pv.x + pv.y;
        ap.y     += pv.z + pv.w;
        at.x     += tv.x + tv.y;
        at.y     += tv.z + tv.w;
    }

    // WMMA wave reductions (EXEC all-1s here: no divergence above)
    float sb = wave_reduce_wmma(abce);
    float si = wave_reduce_wmma(ainter);
    float sp = wave_reduce_wmma(ap);
    float st = wave_reduce_wmma(at);

    __shared__ float lds[TPB1 / 32][4];
    const int lane = threadIdx.x & 31;
    const int wv   = threadIdx.x >> 5;
    if (lane == 0) {
        lds[wv][0] = sb; lds[wv][1] = si; lds[wv][2] = sp; lds[wv][3] = st;
    }
    __syncthreads();
    if (threadIdx.x == 0) {
        float r0 = 0.f, r1 = 0.f, r2 = 0.f, r3 = 0.f;
        for (int w = 0; w < TPB1 / 32; ++w) {
            r0 += lds[w][0]; r1 += lds[w][1]; r2 += lds[w][2]; r3 += lds[w][3];
        }
        float* out = partials + (size_t)blockIdx.x * 4;
        out[0] = r0; out[1] = r1; out[2] = r2; out[3] = r3;
    }
}

// ---------------- Kernel 2: box/conf loss (gather + small reductions) -------
__device__ __forceinline__ float smooth_l1(float x) {
    float d = fabsf(x);
    return (d < 1.0f) ? 0.5f * d * d : d - 0.5f;
}
__device__ __forceinline__ float clampf(float v, float lo, float hi) {
    return fminf(fmaxf(v, lo), hi);
}

__global__ __launch_bounds__(PPTS)
void box_loss_kernel(const float* __restrict__ conf,
                     const float* __restrict__ bbox,
                     const float* __restrict__ tboxes,
                     const int*   __restrict__ bmask,
                     float* __restrict__ boxres) {   // [0..15]=combined [16..31]=conf [32..47]=valid
    const int b  = blockIdx.x;
    const int j  = threadIdx.x;          // 0..319
    const int m  = j / NPTS;
    const int pt = j - m * NPTS;

    const float* tb = tboxes + ((size_t)b * MBOX + m) * 5;
    const float tconf = tb[0];
    const float nx1 = clampf(tb[1] * (1.0f / WW), 0.f, 1.f);
    const float ny1 = clampf(tb[2] * (1.0f / HH), 0.f, 1.f);
    const float nx2 = clampf(tb[3] * (1.0f / WW), 0.f, 1.f);
    const float ny2 = clampf(tb[4] * (1.0f / HH), 0.f, 1.f);

    int x1p = min(max((int)(nx1 * (float)WW), 0), WW - 1);
    int y1p = min(max((int)(ny1 * (float)HH), 0), HH - 1);
    int x2p = min(max((int)(nx2 * (float)WW), 0), WW - 1);
    int y2p = min(max((int)(ny2 * (float)HH), 0), HH - 1);
    int cyy = (y1p + y2p) >> 1;
    int cxx = (x1p + x2p) >> 1;

    int ys, xs;
    switch (pt) {
        case 0: ys = cyy; xs = cxx; break;
        case 1: ys = y1p; xs = x1p; break;
        case 2: ys = y1p; xs = x2p; break;
        case 3: ys = y2p; xs = x1p; break;
        default: ys = y2p; xs = x2p; break;
    }
    const size_t off   = (size_t)ys * WW + xs;
    const size_t plane = (size_t)HH * WW;

    const float cpt = conf[(size_t)b * plane + off];
    const float* bb = bbox + (size_t)b * 4 * plane;
    const float p0 = bb[0 * plane + off];
    const float p1 = bb[1 * plane + off];
    const float p2 = bb[2 * plane + off];
    const float p3 = bb[3 * plane + off];

    const float px1 = clampf(p0, 0.00f, 0.99f);
    const float py1 = clampf(p1, 0.00f, 0.99f);
    const float px2 = clampf(p2, 0.01f, 1.00f);
    const float py2 = clampf(p3, 0.01f, 1.00f);
    const float bx1 = fminf(px1, px2), bx2 = fmaxf(px1, px2);
    const float by1 = fminf(py1, py2), by2 = fmaxf(py1, py2);

    const float pa = (bx2 - bx1) * (by2 - by1);
    const float ta = (nx2 - nx1) * (ny2 - ny1);
    const float ix1 = fmaxf(bx1, nx1), iy1 = fmaxf(by1, ny1);
    const float ix2 = fminf(bx2, nx2), iy2 = fminf(by2, ny2);
    const float inter = fmaxf(ix2 - ix1, 0.f) * fmaxf(iy2 - iy1, 0.f);
    const float uni = pa + ta - inter;
    const float iou = inter / (uni + 1e-6f);
    const float ex1 = fminf(bx1, nx1), ey1 = fminf(by1, ny1);
    const float ex2 = fmaxf(bx2, nx2), ey2 = fmaxf(by2, ny2);
    const float enc = (ex2 - ex1) * (ey2 - ey1);
    const float giou = iou - (enc - uni) / (enc + 1e-6f);

    const float iou_pt  = -__logf(iou + 1e-6f);
    const float giou_pt = 1.0f - giou;
    const float l1 = 0.25f * (smooth_l1(bx1 - nx1) + smooth_l1(by1 - ny1) +
                              smooth_l1(bx2 - nx2) + smooth_l1(by2 - ny2));
    const float maskf = (float)bmask[b * MBOX + m];
    const float cb = bce_elem(cpt, tconf);

    __shared__ float red[5][PPTS];
    red[0][j] = iou_pt  * maskf;
    red[1][j] = l1      * maskf;
    red[2][j] = giou_pt * maskf;
    red[3][j] = cb      * maskf;
    red[4][j] = maskf;
    __syncthreads();

    if (j == 0) {
        float s0 = 0.f, s1 = 0.f, s2 = 0.f, s3 = 0.f, s4 = 0.f;
        for (int i = 0; i < PPTS; ++i) {
            s0 += red[0][i]; s1 += red[1][i]; s2 += red[2][i];
            s3 += red[3][i]; s4 += red[4][i];
        }
        const float cnt = fmaxf(s4, 1.0f);
        boxres[b]      = (0.5f * s0 + 0.3f * s1 + 0.2f * s2) / cnt;  // combined
        boxres[16 + b] = s3 / cnt;                                    // conf per sample
        boxres[32 + b] = (s4 > 0.f) ? 1.0f : 0.0f;                    // valid
    }
}

// ---------------- Kernel 3: deterministic final combine --------------------
__global__ __launch_bounds__(256)
void finalize_kernel(const float* __restrict__ partials,
                     const float* __restrict__ boxres,
                     float* __restrict__ out) {
    __shared__ float lds[4][256];
    const int t = threadIdx.x;
    float a0 = 0.f, a1 = 0.f, a2 = 0.f, a3 = 0.f;
    for (int i = t; i < NBLK1; i += 256) {
        const float* p = partials + (size_t)i * 4;
        a0 += p[0]; a1 += p[1]; a2 += p[2]; a3 += p[3];
    }
    lds[0][t] = a0; lds[1][t] = a1; lds[2][t] = a2; lds[3][t] = a3;
    __syncthreads();
    if (t == 0) {
        float bce = 0.f, pt = 0.f, ps = 0.f, ts = 0.f;
        for (int i = 0; i < 256; ++i) {
            bce += lds[0][i]; pt += lds[1][i]; ps += lds[2][i]; ts += lds[3][i];
        }
        const float bce_mean  = bce / (float)NELEM;
        const float dice      = (2.0f * pt + 1e-5f) / (ps + ts + 1e-5f);
        const float text_loss = 0.5f * bce_mean + 0.5f * (1.0f - dice);

        float sval = 0.f, sbox = 0.f, sconf = 0.f;
        for (int b = 0; b < NB; ++b) {
            const float v = boxres[32 + b];
            sval  += v;
            sbox  += boxres[b] * v;
            sconf += boxres[16 + b] * v;
        }
        const float nvalid    = fmaxf(sval, 1.0f);
        const float box_loss  = sbox / nvalid;
        const float conf_loss = sconf / nvalid;
        out[0] = 1.0f * text_loss + 20.0f * box_loss + 0.5f * conf_loss;
    }
}

// ---------------- launch ----------------------------------------------------
extern "C" void kernel_launch(void* const* d_in, const int* in_sizes, int n_in,
                              void* d_out, int out_size, void* d_ws, size_t ws_size,
                              hipStream_t stream) {
    const float* text_map        = (const float*)d_in[0];
    const float* confidence      = (const float*)d_in[1];
    const float* bbox_coords     = (const float*)d_in[2];
    const float* target_text_map = (const float*)d_in[3];
    const float* target_boxes    = (const float*)d_in[4];
    const int*   box_mask        = (const int*)d_in[5];

    float* ws       = (float*)d_ws;
    float* partials = ws;                 // NBLK1 * 4 floats
    float* boxres   = ws + NBLK1 * 4;     // 48 floats
    float* out      = (float*)d_out;

    text_reduce_kernel<<<NBLK1, TPB1, 0, stream>>>(text_map, target_text_map, partials);
    box_loss_kernel<<<NB, PPTS, 0, stream>>>(confidence, bbox_coords, target_boxes,
                                             box_mask, boxres);
    finalize_kernel<<<1, 256, 0, stream>>>(partials, boxres, out);
}